// Or_Convolution_56564719288842
// MI455X (gfx1250) — compile-verified
//
#include <hip/hip_runtime.h>
#include <stdint.h>

// Problem constants (fixed by the reference's setup_inputs()).
#define N_ROWS  4096
#define ROW_LEN 8192
#define KERNEL_W 16
#define STRIDE_W 4
#define OUT_LEN ((ROW_LEN - KERNEL_W) / STRIDE_W + 1)   // 2045
#define BLOCK   256                                      // 8 wave32 per WG

// ---------------------------------------------------------------------------
// Fast math helpers (lower to v_exp_f32 / v_rcp_f32 on gfx1250)
// ---------------------------------------------------------------------------
__device__ __forceinline__ float fast_exp2(float x) {
#if __has_builtin(__builtin_amdgcn_exp2f)
    return __builtin_amdgcn_exp2f(x);
#else
    return exp2f(x);
#endif
}

__device__ __forceinline__ float fast_rcp(float x) {
#if __has_builtin(__builtin_amdgcn_rcpf)
    return __builtin_amdgcn_rcpf(x);
#else
    return 1.0f / x;
#endif
}

// ---------------------------------------------------------------------------
// gfx1250 async global->LDS copy (ASYNCcnt path), 16 bytes per lane.
// Inline asm, GV mode: vdst = 32-bit LDS byte offset, vaddr = 64-bit global
// address, saddr = off.  (Portable across ROCm-7.2 / amdgpu-toolchain, per
// bridge doc: asm bypasses the toolchain-specific builtin signatures.)
// ---------------------------------------------------------------------------
__device__ __forceinline__ void async_copy_b128(const float* gsrc, float* ldst) {
    uint32_t lds_off = (uint32_t)(uintptr_t)ldst;   // low 32 bits of flat LDS ptr = LDS byte offset
    uint64_t gaddr   = (uint64_t)(uintptr_t)gsrc;
    asm volatile("global_load_async_to_lds_b128 %0, %1, off"
                 :: "v"(lds_off), "v"(gaddr)
                 : "memory");
}

__device__ __forceinline__ void wait_async0() {
#if __has_builtin(__builtin_amdgcn_s_wait_asynccnt)
    __builtin_amdgcn_s_wait_asynccnt(0);
#else
    asm volatile("s_wait_asynccnt 0" ::: "memory");
#endif
}

// ---------------------------------------------------------------------------
// One workgroup per row: stage the 32 KiB row in LDS via async DMA, then each
// thread evaluates strided windows:  out = sum(rho * softmax(rho)) with
// rho = win * w.  Using t = rho*log2e:  out = ln2 * (sum t*2^t) / (sum 2^t).
// ---------------------------------------------------------------------------
__global__ __launch_bounds__(BLOCK) void or_conv_kernel(
    const float* __restrict__ x,     // [N_ROWS, ROW_LEN]
    const float* __restrict__ w,     // [KERNEL_W]
    float* __restrict__ out)         // [N_ROWS, OUT_LEN]
{
    __shared__ __align__(16) float row[ROW_LEN];

    const int tid = threadIdx.x;
    const int r   = blockIdx.x;
    const float* src = x + (size_t)r * ROW_LEN;

    // Stage the row: 8 async B128 chunks per thread, 4 KiB per step, coalesced.
#pragma unroll
    for (int j = 0; j < ROW_LEN / (BLOCK * 4); ++j) {
        const int elt = j * (BLOCK * 4) + tid * 4;     // float index, 16B aligned
        async_copy_b128(src + elt, &row[elt]);
    }
    wait_async0();        // each wave drains its own ASYNCcnt ...
    __syncthreads();      // ... then the barrier makes LDS globally visible

    // Pre-scale weights by log2(e); uniform broadcast load.
    float wl[KERNEL_W];
#pragma unroll
    for (int k = 0; k < KERNEL_W; ++k)
        wl[k] = w[k] * 1.44269504088896340736f;

    const float LN2 = 0.69314718055994530942f;

    for (int o = tid; o < OUT_LEN; o += BLOCK) {
        // 16 floats, four 16B-aligned float4 LDS reads (ds_load_b128).
        const float4* wp = (const float4*)&row[o * STRIDE_W];
        const float4 a = wp[0], b = wp[1], c = wp[2], d = wp[3];

        float t[KERNEL_W];
        t[0]  = a.x * wl[0];  t[1]  = a.y * wl[1];
        t[2]  = a.z * wl[2];  t[3]  = a.w * wl[3];
        t[4]  = b.x * wl[4];  t[5]  = b.y * wl[5];
        t[6]  = b.z * wl[6];  t[7]  = b.w * wl[7];
        t[8]  = c.x * wl[8];  t[9]  = c.y * wl[9];
        t[10] = c.z * wl[10]; t[11] = c.w * wl[11];
        t[12] = d.x * wl[12]; t[13] = d.y * wl[13];
        t[14] = d.z * wl[14]; t[15] = d.w * wl[15];

        // softmax-weighted sum; shift-free (|t| bounded ~6.5, 2^t safe).
        float s = 0.0f, p = 0.0f;
#pragma unroll
        for (int k = 0; k < KERNEL_W; ++k) {
            const float e = fast_exp2(t[k]);           // v_exp_f32
            s += e;
            p = fmaf(t[k], e, p);
        }
        out[(size_t)r * OUT_LEN + o] = p * fast_rcp(s) * LN2;
    }
}

// ---------------------------------------------------------------------------
extern "C" void kernel_launch(void* const* d_in, const int* in_sizes, int n_in,
                              void* d_out, int out_size, void* d_ws, size_t ws_size,
                              hipStream_t stream) {
    (void)in_sizes; (void)n_in; (void)out_size; (void)d_ws; (void)ws_size;
    const float* x = (const float*)d_in[0];   // 4096 x 8192 f32
    const float* w = (const float*)d_in[1];   // 1 x 16 f32
    float* out = (float*)d_out;               // 4096 x 2045 f32

    or_conv_kernel<<<N_ROWS, BLOCK, 0, stream>>>(x, w, out);
}